// GWRouter_87806311400112
// MI455X (gfx1250) — compile-verified
//
#include <hip/hip_runtime.h>
#include <hip/hip_bf16.h>
#include <math.h>

// GWRouter: mean(4x4096x2048 f32) -> 16-way softmax router epilogue.
// Pass 1: grid-wide sum via V_WMMA_F32_16X16X4_F32 with B = ones(4x16)
//         (D[i][j] += sum_k A[i][k]; row-sum replicated over 16 cols).
// Pass 2: single-wave deterministic finalize + router math.

typedef float v2f __attribute__((ext_vector_type(2)));
typedef float v4f __attribute__((ext_vector_type(4)));
typedef float v8f __attribute__((ext_vector_type(8)));

#define NBLOCKS  1024
#define NTHREADS 256
#define NUM_EXPERTS 16

__global__ __launch_bounds__(NTHREADS)
void gw_reduce_wmma(const float* __restrict__ x, float* __restrict__ partials, int n) {
  const int tid      = blockIdx.x * NTHREADS + threadIdx.x;
  const int nthreads = NBLOCKS * NTHREADS;
  const int n4       = n >> 2;
  const int iters    = n4 / nthreads;        // uniform trip count -> EXEC all-ones in WMMA loop
  const int n4_main  = iters * nthreads;

  v8f acc = {};
  const v2f ones = {1.0f, 1.0f};             // B = 4x16 matrix of ones (2 VGPRs, all lanes)
  const v4f* __restrict__ x4 = (const v4f*)x;

  int i = tid;
  for (int it = 0; it < iters; ++it, i += nthreads) {
    v4f v = x4[i];                           // global_load_b128, 512 B per wave per iter
    v2f a0 = {v.x, v.y};
    v2f a1 = {v.z, v.w};
    // D = A(16x4) * ones(4x16) + C  : sums 64 f32 tile elements into accumulator
    acc = __builtin_amdgcn_wmma_f32_16x16x4_f32(false, a0, false, ones, (short)0, acc, false, false);
    acc = __builtin_amdgcn_wmma_f32_16x16x4_f32(false, a1, false, ones, (short)0, acc, false, false);
  }

  // Lane-local sum of the 8 accumulator VGPRs; undo 16x column replication.
  float s = ((acc[0] + acc[1]) + (acc[2] + acc[3])) +
            ((acc[4] + acc[5]) + (acc[6] + acc[7]));
  s *= (1.0f / 16.0f);

  // Remainder (float4s not covered by the uniform loop + n%4 tail): scalar VALU, block 0.
  if (blockIdx.x == 0) {
    for (int j = n4_main * 4 + threadIdx.x; j < n; j += NTHREADS) s += x[j];
  }

  __shared__ float red[NTHREADS];
  red[threadIdx.x] = s;
  __syncthreads();
  for (int off = NTHREADS / 2; off > 0; off >>= 1) {
    if (threadIdx.x < off) red[threadIdx.x] += red[threadIdx.x + off];
    __syncthreads();
  }
  if (threadIdx.x == 0) partials[blockIdx.x] = red[0];
}

__global__ __launch_bounds__(32)
void gw_router_finalize(const float* __restrict__ partials,
                        const float* __restrict__ prototypes,
                        const float* __restrict__ usage_ema,
                        float* __restrict__ out, int n) {
  __shared__ float red[32];
  float s = 0.0f;
  for (int i = threadIdx.x; i < NBLOCKS; i += 32) s += partials[i];  // fixed per-lane order
  red[threadIdx.x] = s;
  __syncthreads();

  if (threadIdx.x == 0) {
    float tot = 0.0f;
    for (int i = 0; i < 32; ++i) tot += red[i];                      // fixed order
    const float xm = tot / (float)n;

    // sim = -(proto - x)^2 ; softmax
    float sim[NUM_EXPERTS], probs[NUM_EXPERTS];
    float m = -INFINITY;
    for (int e = 0; e < NUM_EXPERTS; ++e) {
      float d = prototypes[e] - xm;
      sim[e] = -d * d;
      m = fmaxf(m, sim[e]);
    }
    float z = 0.0f;
    for (int e = 0; e < NUM_EXPERTS; ++e) { probs[e] = expf(sim[e] - m); z += probs[e]; }
    const float inv = 1.0f / z;
    for (int e = 0; e < NUM_EXPERTS; ++e) probs[e] *= inv;

    // top-2, ties -> lowest index (matches lax.top_k)
    int i0 = 0;
    for (int e = 1; e < NUM_EXPERTS; ++e) if (probs[e] > probs[i0]) i0 = e;
    int i1 = (i0 == 0) ? 1 : 0;
    for (int e = 0; e < NUM_EXPERTS; ++e) if (e != i0 && probs[e] > probs[i1]) i1 = e;

    // EMA update + balance loss; write outputs:
    // [0:16) mask, [16:32) probs, [32] loss, [33:49) new_ema, [49:51) topk idx
    const float alpha  = 1.0f / 1000.0f;
    const float target = 1.0f / (float)NUM_EXPERTS;
    float usage_balance = 0.0f;
    for (int e = 0; e < NUM_EXPERTS; ++e) {
      float mk = (e == i0 || e == i1) ? 1.0f : 0.0f;
      float ne = (1.0f - alpha) * usage_ema[e] + alpha * mk;
      out[e]      = mk;
      out[16 + e] = probs[e];
      out[33 + e] = ne;
      float dd = ne - target;
      usage_balance += dd * dd;
    }
    out[32] = (usage_balance / (float)NUM_EXPERTS) * 1e-3f;
    out[49] = (float)i0;
    out[50] = (float)i1;
  }
}

extern "C" void kernel_launch(void* const* d_in, const int* in_sizes, int n_in,
                              void* d_out, int out_size, void* d_ws, size_t ws_size,
                              hipStream_t stream) {
  (void)n_in; (void)out_size; (void)ws_size;
  const float* wm    = (const float*)d_in[0];   // wm_state  [4,4096,2048] f32
  const float* proto = (const float*)d_in[1];   // prototypes [16,1] f32
  const float* ema   = (const float*)d_in[2];   // usage_ema  [16] f32
  float* out         = (float*)d_out;
  float* partials    = (float*)d_ws;            // NBLOCKS floats of scratch
  const int n        = in_sizes[0];

  gw_reduce_wmma<<<NBLOCKS, NTHREADS, 0, stream>>>(wm, partials, n);
  gw_router_finalize<<<1, 32, 0, stream>>>(partials, proto, ema, out, n);
}